// PropertyPredictionMLP_71846212928321
// MI455X (gfx1250) — compile-verified
//
#include <hip/hip_runtime.h>

// Problem constants (match reference).
#define N_NODES 10000
#define E_EDGES 320000
#define SDIM    256
#define VDIM    64

typedef __attribute__((ext_vector_type(2))) float v2f;
typedef __attribute__((ext_vector_type(4))) float v4f;
typedef __attribute__((ext_vector_type(8))) float v8f;

// ---------------------------------------------------------------------------
// Zero-fill workspace region (e_sum + cnt). Required every call: harness only
// poisons once and we must not carry state between calls.
// ---------------------------------------------------------------------------
__global__ void zero_kernel(float* __restrict__ p, int n) {
  int i = blockIdx.x * blockDim.x + threadIdx.x;
  if (i < n) p[i] = 0.0f;
}

// ---------------------------------------------------------------------------
// Edge scatter: e_sum[dst] += e[edge]; cnt[dst] += 1.
// One block per edge, 256 threads = 256 channels. Native f32 global atomics.
// `e` is touched exactly once -> non-temporal load (TH=NT) so the 327 MB
// stream does not evict the hot GEMM working set from L2.
// ---------------------------------------------------------------------------
__global__ __launch_bounds__(SDIM)
void scatter_kernel(const float* __restrict__ e,
                    const long long* __restrict__ dst,
                    float* __restrict__ e_sum,
                    float* __restrict__ cnt) {
  int edge = blockIdx.x;
  int c = threadIdx.x;
  long long d = dst[edge];
  float ev = __builtin_nontemporal_load(&e[(long long)edge * SDIM + c]);
  unsafeAtomicAdd(&e_sum[d * SDIM + c], ev);
  if (c == 0) unsafeAtomicAdd(&cnt[d], 1.0f);
}

// ---------------------------------------------------------------------------
// e_mean = 0.5 * e_sum / max(cnt, 1)   (in place over e_sum)
// ---------------------------------------------------------------------------
__global__ void mean_kernel(float* __restrict__ e_sum,
                            const float* __restrict__ cnt) {
  int i = blockIdx.x * blockDim.x + threadIdx.x;
  if (i < N_NODES * SDIM) {
    int n = i >> 8;  // / SDIM
    e_sum[i] = 0.5f * e_sum[i] * __builtin_amdgcn_rcpf(fmaxf(cnt[n], 1.0f));
  }
}

// Fast silu: x * sigmoid(x) with v_exp_f32 + v_rcp_f32 (no IEEE divide chain).
__device__ __forceinline__ float silu(float x) {
  return x * __builtin_amdgcn_rcpf(1.0f + __expf(-x));
}

// ---------------------------------------------------------------------------
// t[n][o] = silu(s@Ws^T + bs) + silu(e_mean@We^T + be)
// Block = 512 threads = 16 waves. Block handles 16 nodes; wave w handles
// output columns [16w, 16w+16). K=256 swept in steps of 4 with
// V_WMMA_F32_16X16X4_F32. A-tiles (s rows, e_mean rows) staged in LDS
// (padded rows to avoid bank conflicts); B fragments (Ws/We, 512 KB total)
// stream from L2.
// ---------------------------------------------------------------------------
__global__ __launch_bounds__(512)
void gemm_silu_kernel(const float* __restrict__ s,
                      const float* __restrict__ em,
                      const float* __restrict__ Ws,
                      const float* __restrict__ bs,
                      const float* __restrict__ We,
                      const float* __restrict__ be,
                      float* __restrict__ t) {
  __shared__ float As[16][SDIM + 2];  // +2 pad: rows land on different banks
  __shared__ float Ae[16][SDIM + 2];

  const int n0  = blockIdx.x * 16;
  const int tid = threadIdx.x;

  // Cooperative stage of the 16x256 A-tiles.
  for (int i = tid; i < 16 * SDIM; i += 512) {
    int r = i >> 8, c = i & (SDIM - 1);
    As[r][c] = s [(n0 + r) * SDIM + c];
    Ae[r][c] = em[(n0 + r) * SDIM + c];
  }
  __syncthreads();

  const int wave = tid >> 5;
  const int lane = tid & 31;
  const int o0   = wave * 16;
  const int m    = lane & 15;          // A: M row / B: N column / D: N column
  const int koff = (lane >> 4) * 2;    // lanes 16..31 carry K+2,K+3
  const int orow = o0 + m;             // row of Ws/We supplying B[k][o0+m]

  v8f accs = {};
  v8f acce = {};

#pragma unroll 4
  for (int k0 = 0; k0 < SDIM; k0 += 4) {
    v2f a_s = *(const v2f*)&As[m][k0 + koff];
    v2f a_e = *(const v2f*)&Ae[m][k0 + koff];
    v2f b_s = *(const v2f*)&Ws[orow * SDIM + k0 + koff];
    v2f b_e = *(const v2f*)&We[orow * SDIM + k0 + koff];
    // (neg_a, A, neg_b, B, c_mod, C, reuse_a, reuse_b)
    accs = __builtin_amdgcn_wmma_f32_16x16x4_f32(false, a_s, false, b_s,
                                                 (short)0, accs, false, false);
    acce = __builtin_amdgcn_wmma_f32_16x16x4_f32(false, a_e, false, b_e,
                                                 (short)0, acce, false, false);
  }

  const float bias_s = bs[o0 + m];
  const float bias_e = be[o0 + m];
  const int mbase = (lane < 16) ? 0 : 8;  // D layout: VGPR r -> M=r (+8 upper)

#pragma unroll
  for (int r = 0; r < 8; ++r) {
    int mm = r + mbase;
    float xs = silu(accs[r] + bias_s);
    float xe = silu(acce[r] + bias_e);
    t[(n0 + mm) * SDIM + (o0 + m)] = xs + xe;
  }
}

// ---------------------------------------------------------------------------
// out[n] = sum_c (t[n][c] + |dot(v[n][c][:], Wv)|) * Wp[c] + bp
// One block per node; thread c owns channel c. `v` (655 MB, read once) is
// streamed with non-temporal float4 loads — this is the bandwidth-critical
// path that sets the ~130 us roofline. Reduction: wave32 shfl tree + one
// LDS pass over the 8 wave partials (single barrier).
// ---------------------------------------------------------------------------
__global__ __launch_bounds__(SDIM)
void reduce_kernel(const float* __restrict__ v,
                   const float* __restrict__ Wv,
                   const float* __restrict__ Wp,
                   const float* __restrict__ bp,
                   const float* __restrict__ t,
                   float* __restrict__ out) {
  __shared__ float sWv[VDIM];
  __shared__ float wred[SDIM / 32];

  const int n = blockIdx.x;
  const int c = threadIdx.x;

  if (c < VDIM) sWv[c] = Wv[c];
  __syncthreads();

  const v4f* vp = (const v4f*)(v + ((long long)n * SDIM + c) * VDIM);
  float acc = 0.0f;
#pragma unroll
  for (int d4 = 0; d4 < VDIM / 4; ++d4) {
    v4f x = __builtin_nontemporal_load(&vp[d4]);   // TH=NT: read-once stream
    acc += x.x * sWv[4 * d4 + 0] + x.y * sWv[4 * d4 + 1] +
           x.z * sWv[4 * d4 + 2] + x.w * sWv[4 * d4 + 3];
  }

  float val = (t[n * SDIM + c] + fabsf(acc)) * Wp[c];

  // Intra-wave (wave32) reduction.
#pragma unroll
  for (int off = 16; off > 0; off >>= 1)
    val += __shfl_down(val, off, 32);

  const int wave = c >> 5;
  const int lane = c & 31;
  if (lane == 0) wred[wave] = val;
  __syncthreads();

  if (c == 0) {
    float tot = 0.0f;
#pragma unroll
    for (int w = 0; w < SDIM / 32; ++w) tot += wred[w];
    out[n] = tot + bp[0];
  }
}

// ---------------------------------------------------------------------------
// Launcher. Input order (setup_inputs): s, v, p, e, Ws, bs, We, be, Wv, Wp,
// bp, batch, edge_index_global. p/batch unused (p enters only as p.sum()*0).
// ---------------------------------------------------------------------------
extern "C" void kernel_launch(void* const* d_in, const int* in_sizes, int n_in,
                              void* d_out, int out_size, void* d_ws, size_t ws_size,
                              hipStream_t stream) {
  const float* s  = (const float*)d_in[0];
  const float* v  = (const float*)d_in[1];
  const float* e  = (const float*)d_in[3];
  const float* Ws = (const float*)d_in[4];
  const float* bs = (const float*)d_in[5];
  const float* We = (const float*)d_in[6];
  const float* be = (const float*)d_in[7];
  const float* Wv = (const float*)d_in[8];
  const float* Wp = (const float*)d_in[9];
  const float* bp = (const float*)d_in[10];
  const long long* eig = (const long long*)d_in[12];  // [2, E] int64
  const long long* dst = eig + E_EDGES;               // second row

  float* ws    = (float*)d_ws;
  float* e_sum = ws;                       // [N, SDIM] -> becomes e_mean
  float* cnt   = e_sum + N_NODES * SDIM;   // [N]
  float* t     = cnt + N_NODES;            // [N, SDIM]
  float* out   = (float*)d_out;

  const int zn = N_NODES * SDIM + N_NODES;
  zero_kernel<<<(zn + 255) / 256, 256, 0, stream>>>(ws, zn);

  scatter_kernel<<<E_EDGES, SDIM, 0, stream>>>(e, dst, e_sum, cnt);

  mean_kernel<<<(N_NODES * SDIM + 255) / 256, 256, 0, stream>>>(e_sum, cnt);

  gemm_silu_kernel<<<N_NODES / 16, 512, 0, stream>>>(s, e_sum, Ws, bs, We, be, t);

  reduce_kernel<<<N_NODES, SDIM, 0, stream>>>(v, Wv, Wp, bp, t, out);
}